// ILRMA_SR_37546604101779
// MI455X (gfx1250) — compile-verified
//
#include <hip/hip_runtime.h>
#include <cmath>

#define DEV __device__ __forceinline__

constexpr int MI = 1025;   // frequency bins
constexpr int MJ = 2048;   // frames
constexpr int MK = 8;      // NMF bases
constexpr int MN = 2;      // sources == channels
constexpr float EPS = 1e-20f;   // NMF_EPS == IP_EPS == 1e-20

typedef __attribute__((ext_vector_type(2))) float v2f;
typedef __attribute__((ext_vector_type(8))) float v8f;

// D(16x16,f32) = A(16x4,f32) x B(4x16,f32) + C  -- chain two for K=8.
DEV v8f wmma_k4(v2f a, v2f b, v8f c) {
  return __builtin_amdgcn_wmma_f32_16x16x4_f32(false, a, false, b, (short)0, c,
                                               false, false);
}

DEV float frcp(float x) { return __builtin_amdgcn_rcpf(x); }   // v_rcp_f32
DEV float frsq(float x) { return __builtin_amdgcn_rsqf(x); }   // v_rsq_f32
DEV float fsqrt(float x) { return __builtin_amdgcn_sqrtf(x); } // v_sqrt_f32

DEV float2 cmul(float2 a, float2 b) { return make_float2(a.x*b.x - a.y*b.y, a.x*b.y + a.y*b.x); }
DEV float2 cadd(float2 a, float2 b) { return make_float2(a.x + b.x, a.y + b.y); }
DEV float2 csub(float2 a, float2 b) { return make_float2(a.x - b.x, a.y - b.y); }
DEV float2 cscale(float2 a, float s) { return make_float2(a.x*s, a.y*s); }

// X is (M, J, I, 2) float32; return complex X[i, m, j]
DEV float2 loadX(const float* __restrict__ X, int i, int m, int j) {
  int idx = ((m*MJ + j)*MI + i)*2;
  return make_float2(X[idx], X[idx + 1]);
}

// ---------------------------------------------------------------------------
// init: T working copies (per-source row-major), V working copies, W=identity
// T0: (I,K,N)  -> Tw[n][i][k];  V0: (K,J,N) -> Vw[n][k][j];  W: (I,2,2) cplx
// ---------------------------------------------------------------------------
__global__ void k_init(const float* __restrict__ T0, const float* __restrict__ V0,
                       float* __restrict__ Tw, float* __restrict__ Vw,
                       float2* __restrict__ W) {
  int idx = blockIdx.x*blockDim.x + threadIdx.x;
  if (idx < MI*MK*MN) {
    int nn = idx % MN; int k = (idx/MN) % MK; int i = idx/(MN*MK);
    Tw[(nn*MI + i)*MK + k] = T0[idx];
  }
  if (idx < MK*MJ*MN) {
    int nn = idx % MN; int j = (idx/MN) % MJ; int k = idx/(MN*MJ);
    Vw[(nn*MK + k)*MJ + j] = V0[idx];
  }
  if (idx < MI*4) {
    int e = idx & 3;
    W[idx] = make_float2((e == 0 || e == 3) ? 1.0f : 0.0f, 0.0f);
  }
}

// ---------------------------------------------------------------------------
// Y[i,n,j] = sum_m W[i,n,m] * Xc[i,m,j]   (also initial Y with W=I)
// ---------------------------------------------------------------------------
__global__ void k_ymix(const float* __restrict__ X, const float2* __restrict__ W,
                       float2* __restrict__ Y) {
  int idx = blockIdx.x*blockDim.x + threadIdx.x;
  if (idx >= MI*MJ) return;
  int i = idx / MJ, j = idx % MJ;
  float2 x0 = loadX(X, i, 0, j);
  float2 x1 = loadX(X, i, 1, j);
  float2 w00 = W[i*4+0], w01 = W[i*4+1], w10 = W[i*4+2], w11 = W[i*4+3];
  Y[(i*2 + 0)*MJ + j] = cadd(cmul(w00, x0), cmul(w01, x1));
  Y[(i*2 + 1)*MJ + j] = cadd(cmul(w10, x0), cmul(w11, x1));
}

// ---------------------------------------------------------------------------
// NMF T-update, fully on the matrix pipe:
//   R tile       : 2x WMMA  (T stripe x V tile)
//   num/den redux: 8x WMMA  ((P*iR^2 | iR) tile  x  V^T tile chunks)
// One block per 16-row stripe of I; 4 waves each own a j-tile per round.
// T[i,k] = max(T * sqrt(num/(den+eps)), eps)
// ---------------------------------------------------------------------------
__global__ __launch_bounds__(128) void k_nmf_t(const float2* __restrict__ Y,
                                               float* __restrict__ T,
                                               const float* __restrict__ V,
                                               int n) {
  __shared__ float Tst[16][MK];
  __shared__ float stP[4][16][16];
  __shared__ float stI[4][16][16];
  const int t = threadIdx.x;
  const int wave = t >> 5, lane = t & 31;
  const int half = lane >> 4, lcol = lane & 15;
  const int i0 = blockIdx.x * 16;

  for (int idx = t; idx < 16*MK; idx += 128) {
    int r = idx >> 3, k = idx & 7;
    int i = i0 + r;
    Tst[r][k] = (i < MI) ? T[i*MK + k] : 0.0f;
  }
  __syncthreads();

  // A operand of the R GEMM (constant across j): lane holds row i0+lcol,
  // lanes 0-15 K={0,1}, lanes 16-31 K={2,3} (+4 for the second chunk).
  v2f a0, a1;
  a0.x = Tst[lcol][0 + 2*half]; a0.y = Tst[lcol][1 + 2*half];
  a1.x = Tst[lcol][4 + 2*half]; a1.y = Tst[lcol][5 + 2*half];

  v8f accN = {}, accD = {};          // per-wave partials, C layout (i x k)

  for (int r = 0; r < MJ/(16*4); ++r) {      // 32 rounds x 4 waves x 16 = J
    const int j0 = (r*4 + wave)*16;
    const int j = j0 + lcol;
    v2f b0, b1;
    b0.x = V[(0 + 2*half)*MJ + j]; b0.y = V[(1 + 2*half)*MJ + j];
    b1.x = V[(4 + 2*half)*MJ + j]; b1.y = V[(5 + 2*half)*MJ + j];
    v8f c = {};
    c = wmma_k4(a0, b0, c);
    c = wmma_k4(a1, b1, c);
#pragma unroll
    for (int rr = 0; rr < 8; ++rr) {         // c[rr] = R[i0+rr+8*half][j]
      int row = rr + 8*half;
      int i = i0 + row;
      int iy = (i < MI) ? i : (MI - 1);
      float ir = frcp(c[rr] + EPS);
      float2 yv = Y[(iy*MN + n)*MJ + j];
      float p2 = (yv.x*yv.x + yv.y*yv.y) * ir * ir;
      ir = (i < MI) ? ir : 0.0f;
      p2 = (i < MI) ? p2 : 0.0f;
      stP[wave][row][lcol] = p2;
      stI[wave][row][lcol] = ir;
    }
    __syncthreads();
    // num(16i x 8k) += stage(16i x 16j) @ V^T(16j x 8k), as 4 K=4 WMMA chunks.
    // A: row=i=lcol, K=j-chunk (from stage LDS). B: row=j, col=k (V, k<8).
#pragma unroll
    for (int jc = 0; jc < 16; jc += 4) {
      v2f aP, aI, bV;
      aP.x = stP[wave][lcol][jc + 2*half];     aP.y = stP[wave][lcol][jc + 2*half + 1];
      aI.x = stI[wave][lcol][jc + 2*half];     aI.y = stI[wave][lcol][jc + 2*half + 1];
      bV.x = (lcol < MK) ? V[lcol*MJ + j0 + jc + 2*half]     : 0.0f;
      bV.y = (lcol < MK) ? V[lcol*MJ + j0 + jc + 2*half + 1] : 0.0f;
      accN = wmma_k4(aP, bV, accN);
      accD = wmma_k4(aI, bV, accD);
    }
    __syncthreads();
  }
  // cross-wave combine: partials live at (row=i, col=k) in C layout.
#pragma unroll
  for (int rr = 0; rr < 8; ++rr) {
    int row = rr + 8*half;
    stP[wave][row][lcol] = accN[rr];
    stI[wave][row][lcol] = accD[rr];
  }
  __syncthreads();
  const int irow = t >> 3, kk = t & 7;
  int i = i0 + irow;
  if (i < MI) {
    float num = stP[0][irow][kk] + stP[1][irow][kk] + stP[2][irow][kk] + stP[3][irow][kk];
    float den = stI[0][irow][kk] + stI[1][irow][kk] + stI[2][irow][kk] + stI[3][irow][kk];
    float tnew = Tst[irow][kk] * fsqrt(num * frcp(den + EPS));
    T[i*MK + kk] = fmaxf(tnew, EPS);
  }
}

// ---------------------------------------------------------------------------
// NMF V-update: R via WMMA, then num/den(8k x 16j) += T^T chunks x stage tiles
// via WMMA. One block per 16-col tile of J; waves own I-stripes per round.
// V[k,j] = max(V * sqrt(num/(den+eps)), eps)
// ---------------------------------------------------------------------------
__global__ __launch_bounds__(128) void k_nmf_v(const float2* __restrict__ Y,
                                               const float* __restrict__ T,
                                               float* __restrict__ V,
                                               int n) {
  __shared__ float Tst[4][16][MK];
  __shared__ float stP[4][16][16];
  __shared__ float stI[4][16][16];
  const int t = threadIdx.x;
  const int wave = t >> 5, lane = t & 31;
  const int half = lane >> 4, lcol = lane & 15;
  const int j0 = blockIdx.x * 16;
  const int j = j0 + lcol;

  v2f b0, b1;   // B of the R GEMM, constant across the i loop
  b0.x = V[(0 + 2*half)*MJ + j]; b0.y = V[(1 + 2*half)*MJ + j];
  b1.x = V[(4 + 2*half)*MJ + j]; b1.y = V[(5 + 2*half)*MJ + j];

  v8f accN = {}, accD = {};          // per-wave partials, C layout (k x j)

  const int ROUNDS = ((MI + 15)/16 + 3)/4;   // 17
  for (int r = 0; r < ROUNDS; ++r) {
    const int i0 = (r*4 + wave)*16;
#pragma unroll
    for (int q = 0; q < 4; ++q) {            // stage this wave's T stripe
      int idx = lane*4 + q;
      int rr = idx >> 3, k = idx & 7;
      int i = i0 + rr;
      Tst[wave][rr][k] = (i < MI) ? T[i*MK + k] : 0.0f;
    }
    __syncthreads();
    v2f a0, a1;
    a0.x = Tst[wave][lcol][0 + 2*half]; a0.y = Tst[wave][lcol][1 + 2*half];
    a1.x = Tst[wave][lcol][4 + 2*half]; a1.y = Tst[wave][lcol][5 + 2*half];
    v8f c = {};
    c = wmma_k4(a0, b0, c);
    c = wmma_k4(a1, b1, c);
#pragma unroll
    for (int rr = 0; rr < 8; ++rr) {
      int row = rr + 8*half;
      int i = i0 + row;
      int iy = (i < MI) ? i : (MI - 1);
      float ir = frcp(c[rr] + EPS);
      float2 yv = Y[(iy*MN + n)*MJ + j];
      float p2 = (yv.x*yv.x + yv.y*yv.y) * ir * ir;
      ir = (i < MI) ? ir : 0.0f;
      p2 = (i < MI) ? p2 : 0.0f;
      stP[wave][row][lcol] = p2;
      stI[wave][row][lcol] = ir;
    }
    __syncthreads();
    // num(8k x 16j) += T^T(8k x 16i) @ stage(16i x 16j): A row=k (k<8),
    // K=i-chunk from Tst; B row=i, col=j from stage.
#pragma unroll
    for (int ic = 0; ic < 16; ic += 4) {
      v2f aT, bP, bI;
      aT.x = (lcol < MK) ? Tst[wave][ic + 2*half][lcol]     : 0.0f;
      aT.y = (lcol < MK) ? Tst[wave][ic + 2*half + 1][lcol] : 0.0f;
      bP.x = stP[wave][ic + 2*half][lcol];     bP.y = stP[wave][ic + 2*half + 1][lcol];
      bI.x = stI[wave][ic + 2*half][lcol];     bI.y = stI[wave][ic + 2*half + 1][lcol];
      accN = wmma_k4(aT, bP, accN);
      accD = wmma_k4(aT, bI, accD);
    }
    __syncthreads();
  }
  // cross-wave combine: partials at (row=k, col=jj).
#pragma unroll
  for (int rr = 0; rr < 8; ++rr) {
    int row = rr + 8*half;
    stP[wave][row][lcol] = accN[rr];
    stI[wave][row][lcol] = accD[rr];
  }
  __syncthreads();
  const int kk = t >> 4, jj = t & 15;        // kk in [0,8)
  float num = stP[0][kk][jj] + stP[1][kk][jj] + stP[2][kk][jj] + stP[3][kk][jj];
  float den = stI[0][kk][jj] + stI[1][kk][jj] + stI[2][kk][jj] + stI[3][kk][jj];
  float vnew = V[kk*MJ + j0 + jj] * fsqrt(num * frcp(den + EPS));
  V[kk*MJ + j0 + jj] = fmaxf(vnew, EPS);
}

// ---------------------------------------------------------------------------
// Weighted covariance per bin, fused with R=T@V (new T,V) via WMMA:
// D[i] = (1/J) sum_j Xc[i,:,j] Xc[i,:,j]^H / (R[i,j]+eps)
// stored as 4 floats: d00, d11, Re(d01), Im(d01)
// ---------------------------------------------------------------------------
__global__ __launch_bounds__(128) void k_cov(const float* __restrict__ X,
                                             const float* __restrict__ T,
                                             const float* __restrict__ V,
                                             float* __restrict__ D) {
  __shared__ float Tst[16][MK];
  __shared__ float Dacc[16][4];
  const int t = threadIdx.x;
  const int wave = t >> 5, lane = t & 31;
  const int half = lane >> 4, lcol = lane & 15;
  const int i0 = blockIdx.x * 16;

  for (int idx = t; idx < 16*MK; idx += 128) {
    int r = idx >> 3, k = idx & 7;
    int i = i0 + r;
    Tst[r][k] = (i < MI) ? T[i*MK + k] : 0.0f;
  }
  if (t < 64) Dacc[t >> 2][t & 3] = 0.0f;
  __syncthreads();

  v2f a0, a1;
  a0.x = Tst[lcol][0 + 2*half]; a0.y = Tst[lcol][1 + 2*half];
  a1.x = Tst[lcol][4 + 2*half]; a1.y = Tst[lcol][5 + 2*half];

  float s00[8], s11[8], s01r[8], s01i[8];
#pragma unroll
  for (int rr = 0; rr < 8; ++rr) { s00[rr]=0.f; s11[rr]=0.f; s01r[rr]=0.f; s01i[rr]=0.f; }

  for (int r = 0; r < MJ/(16*4); ++r) {
    const int j = (r*4 + wave)*16 + lcol;
    v2f b0, b1;
    b0.x = V[(0 + 2*half)*MJ + j]; b0.y = V[(1 + 2*half)*MJ + j];
    b1.x = V[(4 + 2*half)*MJ + j]; b1.y = V[(5 + 2*half)*MJ + j];
    v8f c = {};
    c = wmma_k4(a0, b0, c);
    c = wmma_k4(a1, b1, c);
#pragma unroll
    for (int rr = 0; rr < 8; ++rr) {
      int i = i0 + rr + 8*half;
      int iy = (i < MI) ? i : (MI - 1);
      float ir = frcp(c[rr] + EPS);
      ir = (i < MI) ? ir : 0.0f;
      float2 x0 = loadX(X, iy, 0, j);
      float2 x1 = loadX(X, iy, 1, j);
      s00[rr]  += (x0.x*x0.x + x0.y*x0.y) * ir;
      s11[rr]  += (x1.x*x1.x + x1.y*x1.y) * ir;
      s01r[rr] += (x0.x*x1.x + x0.y*x1.y) * ir;   // Re(x0 * conj(x1))
      s01i[rr] += (x0.y*x1.x - x0.x*x1.y) * ir;   // Im(x0 * conj(x1))
    }
  }
#pragma unroll
  for (int rr = 0; rr < 8; ++rr) {
    int row = rr + 8*half;
    atomicAdd(&Dacc[row][0], s00[rr]);
    atomicAdd(&Dacc[row][1], s11[rr]);
    atomicAdd(&Dacc[row][2], s01r[rr]);
    atomicAdd(&Dacc[row][3], s01i[rr]);
  }
  __syncthreads();
  if (t < 64) {
    int row = t >> 2, cp = t & 3;
    int i = i0 + row;
    if (i < MI) D[i*4 + cp] = Dacc[row][cp] * (1.0f/(float)MJ);
  }
}

// ---------------------------------------------------------------------------
// Per-bin IP update: A = W@D, b = A^{-1} e_n, w = conj(b)/sqrt(Re(b^H D b)+eps)
// ---------------------------------------------------------------------------
__global__ void k_wupd(const float* __restrict__ D, float2* __restrict__ W, int n) {
  int i = blockIdx.x*blockDim.x + threadIdx.x;
  if (i >= MI) return;
  float d00 = D[i*4+0] + EPS;                         // + IP_EPS * eye
  float d11 = D[i*4+1] + EPS;
  float2 d01 = make_float2(D[i*4+2], D[i*4+3]);
  float2 d10 = make_float2(d01.x, -d01.y);
  float2 w00 = W[i*4+0], w01 = W[i*4+1], w10 = W[i*4+2], w11 = W[i*4+3];
  float2 A00 = cadd(cscale(w00, d00), cmul(w01, d10));
  float2 A01 = cadd(cmul(w00, d01), cscale(w01, d11));
  float2 A10 = cadd(cscale(w10, d00), cmul(w11, d10));
  float2 A11 = cadd(cmul(w10, d01), cscale(w11, d11));
  float2 det = csub(cmul(A00, A11), cmul(A01, A10));
  float idn = frcp(det.x*det.x + det.y*det.y);
  float2 idet = make_float2(det.x*idn, -det.y*idn);
  float2 b0, b1;
  if (n == 0) { b0 = cmul(A11, idet); b1 = cmul(make_float2(-A10.x, -A10.y), idet); }
  else        { b0 = cmul(make_float2(-A01.x, -A01.y), idet); b1 = cmul(A00, idet); }
  float2 Db0 = cadd(cscale(b0, d00), cmul(d01, b1));
  float2 Db1 = cadd(cmul(d10, b0), cscale(b1, d11));
  float quad = b0.x*Db0.x + b0.y*Db0.y + b1.x*Db1.x + b1.y*Db1.y;  // Re(b^H D b)
  float inv = frsq(quad + EPS);
  W[i*4 + n*2 + 0] = make_float2(b0.x*inv, -b0.y*inv);             // conj(b/denom)
  W[i*4 + n*2 + 1] = make_float2(b1.x*inv, -b1.y*inv);
}

// out (N,J,I,2): out[((n*J+j)*I+i)] = Y[i,n,j]  (as float2)
__global__ void k_out(const float2* __restrict__ Y, float2* __restrict__ out) {
  int idx = blockIdx.x*blockDim.x + threadIdx.x;
  if (idx >= MN*MJ*MI) return;
  int i = idx % MI;
  int j = (idx / MI) % MJ;
  int nn = idx / (MI*MJ);
  out[idx] = Y[(i*MN + nn)*MJ + j];
}

extern "C" void kernel_launch(void* const* d_in, const int* in_sizes, int n_in,
                              void* d_out, int out_size, void* d_ws, size_t ws_size,
                              hipStream_t stream) {
  (void)in_sizes; (void)n_in; (void)out_size; (void)ws_size;
  const float* X  = (const float*)d_in[0];
  // d_in[1] == sv, unused by the forward pass
  const float* T0 = (const float*)d_in[2];
  const float* V0 = (const float*)d_in[3];

  char* base = (char*)d_ws;
  size_t off = 0;
  auto take = [&](size_t bytes) -> void* {
    void* p = base + off;
    off += (bytes + 255) & ~(size_t)255;
    return p;
  };
  float2* Y  = (float2*)take(sizeof(float2)*(size_t)MI*MN*MJ);   // 33.6 MB
  float*  Tw = (float*) take(sizeof(float)*(size_t)MN*MI*MK);
  float*  Vw = (float*) take(sizeof(float)*(size_t)MN*MK*MJ);
  float2* W  = (float2*)take(sizeof(float2)*(size_t)MI*4);
  float*  Db = (float*) take(sizeof(float)*(size_t)MI*4);

  const int strips = (MI + 15)/16;            // 65
  const int ymix_blocks = (MI*MJ + 255)/256;

  k_init<<<(MK*MJ*MN + 255)/256, 256, 0, stream>>>(T0, V0, Tw, Vw, W);
  k_ymix<<<ymix_blocks, 256, 0, stream>>>(X, W, Y);   // W == I  =>  Y = Xc

  for (int it = 0; it < 10; ++it) {
    for (int n = 0; n < MN; ++n) {
      float* Tn = Tw + (size_t)n*MI*MK;
      float* Vn = Vw + (size_t)n*MK*MJ;
      k_nmf_t<<<strips, 128, 0, stream>>>(Y, Tn, Vn, n);
      k_nmf_v<<<MJ/16, 128, 0, stream>>>(Y, Tn, Vn, n);
      k_cov<<<strips, 128, 0, stream>>>(X, Tn, Vn, Db);
      k_wupd<<<(MI + 127)/128, 128, 0, stream>>>(Db, W, n);
    }
    k_ymix<<<ymix_blocks, 256, 0, stream>>>(X, W, Y);
  }
  k_out<<<(MN*MJ*MI + 255)/256, 256, 0, stream>>>(Y, (float2*)d_out);
}